// SIEnc_58686433133099
// MI455X (gfx1250) — compile-verified
//
#include <hip/hip_runtime.h>
#include <math.h>

typedef __attribute__((ext_vector_type(2))) float v2f;
typedef __attribute__((ext_vector_type(8))) float v8f;

#define HDIM 64
#define GAT_NEG_SLOPE 0.2f

// ---- order-preserving float <-> uint encoding for atomic max ----
__device__ __forceinline__ unsigned int enc_f32(float f) {
    unsigned int b = __float_as_uint(f);
    return (b & 0x80000000u) ? ~b : (b | 0x80000000u);
}
__device__ __forceinline__ float dec_f32(unsigned int k) {
    unsigned int b = (k & 0x80000000u) ? (k & 0x7fffffffu) : ~k;
    return __uint_as_float(b);
}
#define ENC_NEG_INF 0x007FFFFFu  // enc_f32(-inf)

// ---- wvec[k] = sum_j Wsrc[k][j]*asrc[j] ; wvec[64+k] = sum_j Wdst[k][j]*adst[j] ----
__global__ void gat_wvec_kernel(const float* __restrict__ Wsrc,
                                const float* __restrict__ Wdst,
                                const float* __restrict__ asrc,
                                const float* __restrict__ adst,
                                float* __restrict__ wvec) {
    int k = threadIdx.x;  // 0..63
    float s0 = 0.f, s1 = 0.f;
    for (int j = 0; j < HDIM; ++j) {
        s0 += Wsrc[k * HDIM + j] * asrc[j];
        s1 += Wdst[k * HDIM + j] * adst[j];
    }
    wvec[k] = s0;
    wvec[64 + k] = s1;
}

// ---- WMMA fp32 GEMM: xs = h @ Wsrc ; skip = h @ Wlin + blin ----
// grid: (ceil(N/16), 2), block: 128 (4 waves; wave = one 16-col tile)
__global__ void gat_gemm_wmma(const float* __restrict__ h,
                              const float* __restrict__ Wsrc,
                              const float* __restrict__ Wlin,
                              const float* __restrict__ blin,
                              float* __restrict__ xs,
                              float* __restrict__ skip,
                              int n_nodes) {
    const int lane  = threadIdx.x & 31;
    const int ntile = threadIdx.x >> 5;   // 0..3
    const int row0  = blockIdx.x * 16;
    const int mat   = blockIdx.y;         // 0: Wsrc -> xs, 1: Wlin -> skip(+bias)
    const float* __restrict__ W = (mat == 0) ? Wsrc : Wlin;
    float* __restrict__ out = (mat == 0) ? xs : skip;

    const int half = lane >> 4;           // 0 or 1
    const int mrow = lane & 15;
    const int ncol = ntile * 16 + mrow;

    int arow_idx = row0 + mrow;
    if (arow_idx >= n_nodes) arow_idx = n_nodes - 1;   // clamp (no EXEC divergence)
    const float* __restrict__ arow = h + (size_t)arow_idx * HDIM + half * 2;

    v8f acc = {};
#pragma unroll
    for (int k0 = 0; k0 < HDIM; k0 += 4) {
        v2f a = *(const v2f*)(arow + k0);             // A[M=mrow][k0+2*half .. +1]
        const int kb = k0 + half * 2;
        v2f b;
        b.x = W[(size_t)kb * HDIM + ncol];            // B[kb][ncol]
        b.y = W[(size_t)(kb + 1) * HDIM + ncol];      // B[kb+1][ncol]
        acc = __builtin_amdgcn_wmma_f32_16x16x4_f32(false, a, false, b,
                                                    (short)0, acc, false, false);
    }

    const float bias = (mat == 1) ? blin[ncol] : 0.0f;
#pragma unroll
    for (int r = 0; r < 8; ++r) {
        int row = row0 + r + half * 8;                // C/D layout: VGPR r -> M = r + 8*half
        if (row < n_nodes) out[(size_t)row * HDIM + ncol] = acc[r] + bias;
    }
}

// ---- per-node: a_s, a_d via folded attention vectors; init m, z, agg ----
__global__ void gat_node_prep(const float* __restrict__ h,
                              const float* __restrict__ wvec,
                              float* __restrict__ a_s, float* __restrict__ a_d,
                              unsigned int* __restrict__ mkey,
                              float* __restrict__ z,
                              float* __restrict__ agg, int n_nodes) {
    int n = blockIdx.x * blockDim.x + threadIdx.x;
    if (n >= n_nodes) return;
    const float* __restrict__ row = h + (size_t)n * HDIM;
    float s0 = 0.f, s1 = 0.f;
#pragma unroll 8
    for (int k = 0; k < HDIM; ++k) {
        float v = row[k];
        s0 += v * wvec[k];
        s1 += v * wvec[64 + k];
    }
    a_s[n] = s0;
    a_d[n] = s1;
    mkey[n] = ENC_NEG_INF;
    z[n] = 0.f;
    float4* ag = (float4*)(agg + (size_t)n * HDIM);
#pragma unroll
    for (int i = 0; i < HDIM / 4; ++i) ag[i] = make_float4(0.f, 0.f, 0.f, 0.f);
}

// ---- per-edge: e = leaky_relu(a_s[src]+a_d[dst]); segment max over dst ----
__global__ void gat_edge_max(const int* __restrict__ src, const int* __restrict__ dst,
                             const float* __restrict__ a_s, const float* __restrict__ a_d,
                             float* __restrict__ ebuf, unsigned int* __restrict__ mkey,
                             int n_edges) {
    int i = blockIdx.x * blockDim.x + threadIdx.x;
    if (i >= n_edges) return;
    int s = src[i], d = dst[i];
    float e = a_s[s] + a_d[d];
    e = (e > 0.f) ? e : GAT_NEG_SLOPE * e;
    ebuf[i] = e;
    atomicMax(&mkey[d], enc_f32(e));
}

// ---- per-edge: ee = exp(e - m[dst]); segment sum z ----
__global__ void gat_edge_exp(const int* __restrict__ dst,
                             const unsigned int* __restrict__ mkey,
                             float* __restrict__ ebuf, float* __restrict__ z,
                             int n_edges) {
    int i = blockIdx.x * blockDim.x + threadIdx.x;
    if (i >= n_edges) return;
    int d = dst[i];
    float ee = __expf(ebuf[i] - dec_f32(mkey[d]));
    ebuf[i] = ee;
    atomicAdd(&z[d], ee);
}

// ---- per-edge scatter: agg[dst] += alpha * xs[src]; one wave32 per edge ----
__global__ void gat_edge_scatter(const int* __restrict__ src, const int* __restrict__ dst,
                                 const float* __restrict__ ebuf, const float* __restrict__ z,
                                 const float* __restrict__ xs, float* __restrict__ agg,
                                 int n_edges) {
    int lane = threadIdx.x & 31;
    int i = blockIdx.x * (blockDim.x >> 5) + (threadIdx.x >> 5);
    if (i >= n_edges) return;
    int s = src[i], d = dst[i];
    float alpha = ebuf[i] / (z[d] + 1e-16f);
    const float* __restrict__ xrow = xs + (size_t)s * HDIM;
    float* __restrict__ arow = agg + (size_t)d * HDIM;
    atomicAdd(&arow[lane], alpha * xrow[lane]);
    atomicAdd(&arow[lane + 32], alpha * xrow[lane + 32]);
}

// ---- hout = relu(agg + bconv + skip) ----
__global__ void gat_finalize(const float* __restrict__ agg, const float* __restrict__ skip,
                             const float* __restrict__ bconv, float* __restrict__ hout,
                             int total) {
    int i = blockIdx.x * blockDim.x + threadIdx.x;
    if (i >= total) return;
    int c = i & (HDIM - 1);
    float v = agg[i] + bconv[c] + skip[i];
    hout[i] = (v > 0.f) ? v : 0.f;
}

extern "C" void kernel_launch(void* const* d_in, const int* in_sizes, int n_in,
                              void* d_out, int out_size, void* d_ws, size_t ws_size,
                              hipStream_t stream) {
    const float* x     = (const float*)d_in[0];
    const int*   ei    = (const int*)d_in[1];
    const float* Wsrc  = (const float*)d_in[2];
    const float* Wdst  = (const float*)d_in[3];
    const float* asrc  = (const float*)d_in[4];
    const float* adst  = (const float*)d_in[5];
    const float* bconv = (const float*)d_in[6];
    const float* Wlin  = (const float*)d_in[7];
    const float* blin  = (const float*)d_in[8];
    float* out = (float*)d_out;

    const int n_nodes = in_sizes[0] / HDIM;
    const int n_edges = in_sizes[1] / 2;
    const int* src = ei;
    const int* dst = ei + n_edges;

    // workspace carve-up (floats)
    float* ws = (float*)d_ws;
    const size_t nh = (size_t)n_nodes * HDIM;
    float* h1   = ws; ws += nh;
    float* xs   = ws; ws += nh;
    float* skip = ws; ws += nh;
    float* agg  = ws; ws += nh;
    float* a_s  = ws; ws += n_nodes;
    float* a_d  = ws; ws += n_nodes;
    unsigned int* mkey = (unsigned int*)ws; ws += n_nodes;
    float* z    = ws; ws += n_nodes;
    float* ebuf = ws; ws += n_edges;
    float* wvec = ws; ws += 128;

    const float* hin = x;
    for (int l = 0; l < 2; ++l) {
        const float* Ws  = Wsrc  + (size_t)l * HDIM * HDIM;
        const float* Wd  = Wdst  + (size_t)l * HDIM * HDIM;
        const float* Wl  = Wlin  + (size_t)l * HDIM * HDIM;
        const float* as_ = asrc  + (size_t)l * HDIM;
        const float* ad_ = adst  + (size_t)l * HDIM;
        const float* bc  = bconv + (size_t)l * HDIM;
        const float* bl  = blin  + (size_t)l * HDIM;
        float* hout = (l == 0) ? h1 : out;

        gat_wvec_kernel<<<1, 64, 0, stream>>>(Ws, Wd, as_, ad_, wvec);

        dim3 ggrid((n_nodes + 15) / 16, 2);
        gat_gemm_wmma<<<ggrid, 128, 0, stream>>>(hin, Ws, Wl, bl, xs, skip, n_nodes);

        gat_node_prep<<<(n_nodes + 255) / 256, 256, 0, stream>>>(hin, wvec, a_s, a_d,
                                                                 mkey, z, agg, n_nodes);
        gat_edge_max<<<(n_edges + 255) / 256, 256, 0, stream>>>(src, dst, a_s, a_d,
                                                                ebuf, mkey, n_edges);
        gat_edge_exp<<<(n_edges + 255) / 256, 256, 0, stream>>>(dst, mkey, ebuf, z, n_edges);
        gat_edge_scatter<<<(n_edges + 7) / 8, 256, 0, stream>>>(src, dst, ebuf, z,
                                                                xs, agg, n_edges);
        gat_finalize<<<((int)nh + 255) / 256, 256, 0, stream>>>(agg, skip, bc, hout, (int)nh);
        hin = hout;
    }
}